// DHYPRLayer_15745350107692
// MI455X (gfx1250) — compile-verified
//
#include <hip/hip_runtime.h>
#include <hip/hip_bf16.h>

#define D 256
#define MIN_NORM 1e-15f
#define MAXNORM  0.996f   // (1 - 4e-3) / sqrt(c), c = 1

typedef __attribute__((ext_vector_type(2))) float v2f;
typedef __attribute__((ext_vector_type(8))) float v8f;

__device__ __forceinline__ float artanh_clip(float x) {
    x = fminf(fmaxf(x, -1.0f + 1e-7f), 1.0f - 1e-7f);
    return 0.5f * (log1pf(x) - log1pf(-x));
}

// blockDim.x == 256 assumed
__device__ __forceinline__ float block_reduce_sum(float v, float* sdata) {
    int tid = threadIdx.x;
    sdata[tid] = v;
    __syncthreads();
    #pragma unroll
    for (int s = 128; s > 0; s >>= 1) {
        if (tid < s) sdata[tid] += sdata[tid + s];
        __syncthreads();
    }
    float r = sdata[0];
    __syncthreads();
    return r;
}

// h0 = proj(expmap0(x))  — both ops are scalar multiples of the row
__global__ void encode_kernel(const float* __restrict__ x, float* __restrict__ h0) {
    __shared__ float sdata[256];
    int row = blockIdx.x, tid = threadIdx.x;
    float v = x[(size_t)row * D + tid];
    float sq = block_reduce_sum(v * v, sdata);
    float un = fmaxf(sqrtf(sq), MIN_NORM);
    float coef = tanhf(un) / un;           // expmap0 scale
    float n = coef * sqrtf(sq);
    float s = (fmaxf(n, MIN_NORM) > MAXNORM) ? (MAXNORM / n) : 1.0f;
    h0[(size_t)row * D + tid] = s * coef * v;
}

// hypb[0..255] = proj(expmap0(b)), hypb[256] = ||hypb||^2
__global__ void hyp_bias_kernel(const float* __restrict__ b, float* __restrict__ hypb) {
    __shared__ float sdata[256];
    int tid = threadIdx.x;
    float v = b[tid];
    float sq = block_reduce_sum(v * v, sdata);
    float un = fmaxf(sqrtf(sq), MIN_NORM);
    float coef = tanhf(un) / un;
    float n = coef * sqrtf(sq);
    float s = (fmaxf(n, MIN_NORM) > MAXNORM) ? (MAXNORM / n) : 1.0f;
    float g = s * coef;
    hypb[tid] = g * v;
    if (tid == 0) hypb[D] = g * g * sq;
}

// C = A @ W^T : one wave per 16x16 tile, f32 WMMA, K stepped by 4.
// A: [N][256] row-major (h).  W: [256][256] row-major, B = W^T so B[k][n] = W[n*256+k].
__global__ void gemm_wmma_kernel(const float* __restrict__ A,
                                 const float* __restrict__ W,
                                 float* __restrict__ Cout, int nTiles) {
    int wave = blockIdx.x * (blockDim.x >> 5) + (threadIdx.x >> 5);
    if (wave >= nTiles) return;              // wave-uniform: EXEC stays all-ones
    int tileM = wave >> 4;                   // D/16 == 16 N-tiles
    int tileN = wave & 15;
    int lane = threadIdx.x & 31;
    int half = lane >> 4;                    // 0: K{0,1}, 1: K{2,3}
    int l    = lane & 15;
    const float* arow = A + (size_t)(tileM * 16 + l) * D;   // M = l
    const float* brow = W + (size_t)(tileN * 16 + l) * D;   // N = l (row of W = col of B)
    v8f acc = {};
    #pragma unroll 8
    for (int k = 0; k < D; k += 4) {
        v2f a, b;
        a.x = arow[k + half * 2 + 0];
        a.y = arow[k + half * 2 + 1];
        b.x = brow[k + half * 2 + 0];
        b.y = brow[k + half * 2 + 1];
        acc = __builtin_amdgcn_wmma_f32_16x16x4_f32(
            false, a, false, b, (short)0, acc, false, false);
    }
    // C/D layout: VGPR r -> M = r + half*8, N = l
    float* ccol = Cout + (size_t)tileN * 16 + l;
    #pragma unroll
    for (int r = 0; r < 8; ++r) {
        ccol[(size_t)(tileM * 16 + r + half * 8) * D] = acc[r];
    }
}

// t = logmap0( proj( mobius_add( proj(mobius_matvec_res), hyp_b ) ) )
// All proj/expmap/logmap steps are scalar multiples -> 4 reductions total.
__global__ void linear_post_kernel(const float* __restrict__ h_in,
                                   const float* __restrict__ mx,
                                   const float* __restrict__ hypb,
                                   float* __restrict__ t) {
    __shared__ float sdata[256];
    int row = blockIdx.x, tid = threadIdx.x;
    float hv = h_in[(size_t)row * D + tid];
    float mv = mx[(size_t)row * D + tid];
    float bv = hypb[tid];

    float hsq   = block_reduce_sum(hv * hv, sdata);
    float msq   = block_reduce_sum(mv * mv, sdata);
    float mdotb = block_reduce_sum(mv * bv, sdata);

    float xn      = fmaxf(sqrtf(hsq), MIN_NORM);
    float mxn_raw = sqrtf(msq);
    float mxn     = fmaxf(mxn_raw, MIN_NORM);
    float g = tanhf((mxn / xn) * artanh_clip(xn)) / mxn;   // sc = 1
    if (msq == 0.0f) g = 0.0f;                             // all-zero mx row
    // proj(res), res = g*mx -> scalar rescale of g
    float n1 = fmaxf(g * mxn_raw, MIN_NORM);
    if (n1 > MAXNORM) g *= MAXNORM / n1;
    // mobius_add(g*mx, hypb), c = 1
    float x2 = g * g * msq;
    float y2 = hypb[D];
    float xy = g * mdotb;
    float Af  = 1.0f + 2.0f * xy + y2;
    float Bf  = 1.0f - x2;
    float den = fmaxf(1.0f + 2.0f * xy + x2 * y2, MIN_NORM);
    float outv = (Af * g * mv + Bf * bv) / den;
    // proj then logmap0 (scalar on out)
    float osq = block_reduce_sum(outv * outv, sdata);
    float n2 = fmaxf(sqrtf(osq), MIN_NORM);
    float s2 = (n2 > MAXNORM) ? (MAXNORM / n2) : 1.0f;
    float pn = fmaxf(s2 * n2, MIN_NORM);
    float lc = artanh_clip(pn) / pn * s2;
    t[(size_t)row * D + tid] = lc * outv;
}

__global__ void zero_kernel(float* __restrict__ p, int n) {
    int i = blockIdx.x * blockDim.x + threadIdx.x;
    if (i < n) p[i] = 0.0f;
}

// agg[dst] += w * t[src] ; thread = feature lane (coalesced), 8 edges per block
__global__ void spmm_kernel(const float* __restrict__ t,
                            const long long* __restrict__ ei,  // [2][E] flat, int64
                            const float* __restrict__ ew,
                            float* __restrict__ agg, int E) {
    int tid = threadIdx.x;
    int e0 = blockIdx.x * 8;
    #pragma unroll
    for (int i = 0; i < 8; ++i) {
        int e = e0 + i;
        if (e >= E) return;
        int s = (int)ei[e];
        int d = (int)ei[(size_t)E + e];
        float w = ew[e];
        atomicAdd(&agg[(size_t)d * D + tid], w * t[(size_t)s * D + tid]);
    }
}

// emb = proj(expmap0( relu( logmap0( proj(expmap0(agg)) ) ) ))  — 2 reductions
__global__ void agg_post_kernel(const float* __restrict__ agg, float* __restrict__ emb) {
    __shared__ float sdata[256];
    int row = blockIdx.x, tid = threadIdx.x;
    float av = agg[(size_t)row * D + tid];
    float asq = block_reduce_sum(av * av, sdata);
    float un = fmaxf(sqrtf(asq), MIN_NORM);
    float ge = tanhf(un) / un;                 // expmap0 scale
    float ne = ge * sqrtf(asq);
    float s = (fmaxf(ne, MIN_NORM) > MAXNORM) ? (MAXNORM / ne) : 1.0f;
    float gh = ge * s;                         // total scale after proj
    float pn = fmaxf(s * ne, MIN_NORM);
    float lc = artanh_clip(pn) / pn * gh;      // logmap0 total scale on agg
    float xt = fmaxf(lc * av, 0.0f);           // relu in tangent space
    float xsq = block_reduce_sum(xt * xt, sdata);
    float un2 = fmaxf(sqrtf(xsq), MIN_NORM);
    float c2 = tanhf(un2) / un2;
    float n3 = c2 * sqrtf(xsq);
    float s3 = (fmaxf(n3, MIN_NORM) > MAXNORM) ? (MAXNORM / n3) : 1.0f;
    emb[(size_t)row * D + tid] = s3 * c2 * xt;
}

extern "C" void kernel_launch(void* const* d_in, const int* in_sizes, int n_in,
                              void* d_out, int out_size, void* d_ws, size_t ws_size,
                              hipStream_t stream) {
    const float*     x  = (const float*)d_in[0];
    const float*     W1 = (const float*)d_in[1];
    const float*     b1 = (const float*)d_in[2];
    const float*     W2 = (const float*)d_in[3];
    const float*     b2 = (const float*)d_in[4];
    const long long* ei = (const long long*)d_in[5];   // int64 [2, E]
    const float*     ew = (const float*)d_in[6];

    const int N = in_sizes[0] / D;     // 10000
    const int E = in_sizes[6];         // 320000
    float* out = (float*)d_out;        // [2, N, D]

    float* mx   = (float*)d_ws;                  // N*D
    float* tbuf = mx   + (size_t)N * D;          // N*D
    float* agg  = tbuf + (size_t)N * D;          // N*D
    float* h0   = agg  + (size_t)N * D;          // N*D
    float* hypb = h0   + (size_t)N * D;          // D+1

    encode_kernel<<<N, 256, 0, stream>>>(x, h0);

    const int mTiles = (N + 15) / 16;            // 625 (exact, N % 16 == 0)
    const int nTiles = mTiles * (D / 16);        // 10000 tiles, one wave each

    for (int layer = 0; layer < 2; ++layer) {
        const float* W    = layer ? W2 : W1;
        const float* b    = layer ? b2 : b1;
        const float* h_in = layer ? out : h0;    // layer 2 input = emb1 (in d_out)
        float* emb        = out + (size_t)layer * N * D;

        hyp_bias_kernel<<<1, 256, 0, stream>>>(b, hypb);
        gemm_wmma_kernel<<<(nTiles + 3) / 4, 128, 0, stream>>>(h_in, W, mx, nTiles);
        linear_post_kernel<<<N, 256, 0, stream>>>(h_in, mx, hypb, tbuf);
        zero_kernel<<<(N * D + 255) / 256, 256, 0, stream>>>(agg, N * D);
        spmm_kernel<<<(E + 7) / 8, 256, 0, stream>>>(tbuf, ei, ew, agg, E);
        agg_post_kernel<<<N, 256, 0, stream>>>(agg, emb);
    }
}